// LipsNet_44633300140660
// MI455X (gfx1250) — compile-verified
//
#include <hip/hip_runtime.h>
#include <math.h>

typedef __attribute__((ext_vector_type(2))) float v2f;
typedef __attribute__((ext_vector_type(8))) float v8f;

// D = A(16x4,f32) x B(4x16,f32) + C(16x16,f32)  -> v_wmma_f32_16x16x4_f32
__device__ __forceinline__ v8f wmma_f32(v2f a, v2f b, v8f c) {
    return __builtin_amdgcn_wmma_f32_16x16x4_f32(false, a, false, b, (short)0, c,
                                                 false, false);
}

// ---------------------------------------------------------------------------
// C[M,N] = act(A[M,K] @ W[N,K]^T + bias[N]);  MODE 0 = tanh, 1 = identity.
// One wave per 64(M) x 32(N) strip: 8 accumulators (4 row x 2 col tiles).
// Per K-step: 4x b64 A-loads + 2x b64 B-loads feed 8 WMMAs.
// ---------------------------------------------------------------------------
template <int MODE>
__global__ void __launch_bounds__(32)
fwd_gemm(const float* __restrict__ A, const float* __restrict__ W,
         const float* __restrict__ bias, float* __restrict__ H,
         int M, int N, int K) {
    const int ntn  = N >> 5;             // 32-col blocks
    const int rb   = blockIdx.x / ntn;   // 64-row block
    const int nb   = blockIdx.x % ntn;
    const int lane = threadIdx.x & 31;
    const int half = lane >> 4;
    const int lr   = lane & 15;
    const int colA = (nb << 5) + lr;
    const int colB = colA + 16;
    const int row0 = (rb << 6) + lr;

    const float* a0 = A + (size_t)(row0)      * K;
    const float* a1 = A + (size_t)(row0 + 16) * K;
    const float* a2 = A + (size_t)(row0 + 32) * K;
    const float* a3 = A + (size_t)(row0 + 48) * K;
    const float* wA = W + (size_t)colA * K;
    const float* wB = W + (size_t)colB * K;

    v8f c00 = {}, c10 = {}, c20 = {}, c30 = {};
    v8f c01 = {}, c11 = {}, c21 = {}, c31 = {};
    for (int k = 0; k < K; k += 4) {
        const int kk = k + 2 * half;
        v2f bA = *(const v2f*)(wA + kk);
        v2f bB = *(const v2f*)(wB + kk);
        v2f f0 = *(const v2f*)(a0 + kk);
        v2f f1 = *(const v2f*)(a1 + kk);
        v2f f2 = *(const v2f*)(a2 + kk);
        v2f f3 = *(const v2f*)(a3 + kk);
        c00 = wmma_f32(f0, bA, c00);
        c10 = wmma_f32(f1, bA, c10);
        c20 = wmma_f32(f2, bA, c20);
        c30 = wmma_f32(f3, bA, c30);
        c01 = wmma_f32(f0, bB, c01);
        c11 = wmma_f32(f1, bB, c11);
        c21 = wmma_f32(f2, bB, c21);
        c31 = wmma_f32(f3, bB, c31);
    }

    const float bnA = bias[colA];
    const float bnB = bias[colB];
#define EPILOG(ACC, Q, COL, BN)                                               \
    _Pragma("unroll")                                                         \
    for (int v = 0; v < 8; ++v) {                                             \
        float xv = ACC[v] + (BN);                                             \
        if (MODE == 0) xv = tanhf(xv);                                        \
        H[(size_t)((rb << 6) + (Q << 4) + v + 8 * half) * N + (COL)] = xv;    \
    }
    EPILOG(c00, 0, colA, bnA) EPILOG(c10, 1, colA, bnA)
    EPILOG(c20, 2, colA, bnA) EPILOG(c30, 3, colA, bnA)
    EPILOG(c01, 0, colB, bnB) EPILOG(c11, 1, colB, bnB)
    EPILOG(c21, 2, colB, bnB) EPILOG(c31, 3, colB, bnB)
#undef EPILOG
}

// ---------------------------------------------------------------------------
// Per-sample Jacobian Frobenius norm^2.  One workgroup (8 waves) per sample.
//   d1 = 1-h1^2, d2 = 1-h2^2 (LDS)
//   Stage A: Ms[64,512] = W3 @ (diag(d2) W2)   (diag folded into B fragments)
//   Stage B: J[64,256]  = Ms @ (diag(d1) W1)   -> sum(J^2) tree-reduced
// Each wave: 4 row-tiles x 2 col-tiles = 8 accumulators; A fragments shared
// across both column tiles, so 8 VMEM ops feed 8 WMMAs per K-step.
// ---------------------------------------------------------------------------
__global__ void __launch_bounds__(256)
jac_norm(const float* __restrict__ h1, const float* __restrict__ h2,
         const float* __restrict__ W1, const float* __restrict__ W2,
         const float* __restrict__ W3, float* __restrict__ normsq) {
    extern __shared__ float lds[];
    float* Ms  = lds;             // 64*512  (128 KB)
    float* d1s = Ms + 64 * 512;   // 512
    float* d2s = d1s + 512;       // 512
    float* red = d2s + 512;       // 256

    const int b   = blockIdx.x;
    const int tid = threadIdx.x;

    for (int t = tid; t < 512; t += 256) {
        float a = h1[(size_t)b * 512 + t];
        float c = h2[(size_t)b * 512 + t];
        d1s[t] = 1.0f - a * a;
        d2s[t] = 1.0f - c * c;
    }
    __syncthreads();

    const int wave = tid >> 5;
    const int lane = tid & 31;
    const int half = lane >> 4;
    const int lr   = lane & 15;

    const float* w3r0 = W3 + (size_t)(lr)      * 512;
    const float* w3r1 = W3 + (size_t)(lr + 16) * 512;
    const float* w3r2 = W3 + (size_t)(lr + 32) * 512;
    const float* w3r3 = W3 + (size_t)(lr + 48) * 512;

    // ---- Stage A: 32 col-tiles; wave covers {w, w+8} then {w+16, w+24} ----
    for (int p = 0; p < 2; ++p) {
        const int colA = ((wave + p * 16) << 4) + lr;
        const int colB = colA + 128;            // +8 tiles
        v8f c00 = {}, c10 = {}, c20 = {}, c30 = {};
        v8f c01 = {}, c11 = {}, c21 = {}, c31 = {};
        for (int k = 0; k < 512; k += 4) {
            const int kk = k + 2 * half;
            v2f d = *(const v2f*)(d2s + kk);
            v2f bA;
            bA.x = W2[(size_t)kk * 512 + colA];
            bA.y = W2[(size_t)(kk + 1) * 512 + colA];
            bA *= d;
            v2f bB;
            bB.x = W2[(size_t)kk * 512 + colB];
            bB.y = W2[(size_t)(kk + 1) * 512 + colB];
            bB *= d;
            v2f f0 = *(const v2f*)(w3r0 + kk);
            v2f f1 = *(const v2f*)(w3r1 + kk);
            v2f f2 = *(const v2f*)(w3r2 + kk);
            v2f f3 = *(const v2f*)(w3r3 + kk);
            c00 = wmma_f32(f0, bA, c00);
            c10 = wmma_f32(f1, bA, c10);
            c20 = wmma_f32(f2, bA, c20);
            c30 = wmma_f32(f3, bA, c30);
            c01 = wmma_f32(f0, bB, c01);
            c11 = wmma_f32(f1, bB, c11);
            c21 = wmma_f32(f2, bB, c21);
            c31 = wmma_f32(f3, bB, c31);
        }
#define STORE_MS(ACC, Q, COL)                                                 \
        _Pragma("unroll")                                                     \
        for (int v = 0; v < 8; ++v)                                           \
            Ms[(size_t)((Q << 4) + v + 8 * half) * 512 + (COL)] = ACC[v];
        STORE_MS(c00, 0, colA) STORE_MS(c10, 1, colA)
        STORE_MS(c20, 2, colA) STORE_MS(c30, 3, colA)
        STORE_MS(c01, 0, colB) STORE_MS(c11, 1, colB)
        STORE_MS(c21, 2, colB) STORE_MS(c31, 3, colB)
#undef STORE_MS
    }
    __syncthreads();

    // ---- Stage B: 16 col-tiles; wave covers {w, w+8} in one pass ----------
    float sumsq = 0.0f;
    {
        const int colA = (wave << 4) + lr;
        const int colB = colA + 128;            // +8 tiles
        const float* ms0 = Ms + (size_t)(lr)      * 512;
        const float* ms1 = Ms + (size_t)(lr + 16) * 512;
        const float* ms2 = Ms + (size_t)(lr + 32) * 512;
        const float* ms3 = Ms + (size_t)(lr + 48) * 512;

        v8f c00 = {}, c10 = {}, c20 = {}, c30 = {};
        v8f c01 = {}, c11 = {}, c21 = {}, c31 = {};
        for (int k = 0; k < 512; k += 4) {
            const int kk = k + 2 * half;
            v2f d = *(const v2f*)(d1s + kk);
            v2f bA;
            bA.x = W1[(size_t)kk * 256 + colA];
            bA.y = W1[(size_t)(kk + 1) * 256 + colA];
            bA *= d;
            v2f bB;
            bB.x = W1[(size_t)kk * 256 + colB];
            bB.y = W1[(size_t)(kk + 1) * 256 + colB];
            bB *= d;
            v2f f0 = *(const v2f*)(ms0 + kk);
            v2f f1 = *(const v2f*)(ms1 + kk);
            v2f f2 = *(const v2f*)(ms2 + kk);
            v2f f3 = *(const v2f*)(ms3 + kk);
            c00 = wmma_f32(f0, bA, c00);
            c10 = wmma_f32(f1, bA, c10);
            c20 = wmma_f32(f2, bA, c20);
            c30 = wmma_f32(f3, bA, c30);
            c01 = wmma_f32(f0, bB, c01);
            c11 = wmma_f32(f1, bB, c11);
            c21 = wmma_f32(f2, bB, c21);
            c31 = wmma_f32(f3, bB, c31);
        }
#pragma unroll
        for (int v = 0; v < 8; ++v)
            sumsq += c00[v] * c00[v] + c10[v] * c10[v] +
                     c20[v] * c20[v] + c30[v] * c30[v] +
                     c01[v] * c01[v] + c11[v] * c11[v] +
                     c21[v] * c21[v] + c31[v] * c31[v];
    }

    // deterministic tree reduction
    red[tid] = sumsq;
    __syncthreads();
    for (int s = 128; s > 0; s >>= 1) {
        if (tid < s) red[tid] += red[tid + s];
        __syncthreads();
    }
    if (tid == 0) normsq[b] = red[0];
}

// ---------------------------------------------------------------------------
// out = softplus(K) * f / (sqrt(normsq[b]) + eps)
// ---------------------------------------------------------------------------
__global__ void __launch_bounds__(256)
finalize(const float* __restrict__ f, const float* __restrict__ normsq,
         const float* __restrict__ Kp, float* __restrict__ out, int n) {
    int i = blockIdx.x * blockDim.x + threadIdx.x;
    if (i >= n) return;
    float k    = Kp[0];
    float soft = (k > 20.0f) ? k : log1pf(expf(k));  // overflow-safe softplus
    int   b    = i >> 6;                              // Dout = 64
    float nrm  = sqrtf(normsq[b]);
    out[i] = soft * f[i] / (nrm + 1e-5f);
}

extern "C" void kernel_launch(void* const* d_in, const int* in_sizes, int n_in,
                              void* d_out, int out_size, void* d_ws, size_t ws_size,
                              hipStream_t stream) {
    (void)in_sizes; (void)n_in; (void)out_size; (void)ws_size;
    const float* x  = (const float*)d_in[0];
    const float* W1 = (const float*)d_in[1];
    const float* b1 = (const float*)d_in[2];
    const float* W2 = (const float*)d_in[3];
    const float* b2 = (const float*)d_in[4];
    const float* W3 = (const float*)d_in[5];
    const float* b3 = (const float*)d_in[6];
    const float* Kp = (const float*)d_in[7];
    float*       out = (float*)d_out;

    const int B = 2048, Din = 256, H = 512, Dout = 64;

    float* h1 = (float*)d_ws;                 // [B,H]
    float* h2 = h1 + (size_t)B * H;           // [B,H]
    float* f  = h2 + (size_t)B * H;           // [B,Dout]
    float* ns = f + (size_t)B * Dout;         // [B]

    // forward: h1 = tanh(x@W1^T+b1); h2 = tanh(h1@W2^T+b2); f = h2@W3^T+b3
    fwd_gemm<0><<<(B / 64) * (H / 32), 32, 0, stream>>>(x, W1, b1, h1, B, H, Din);
    fwd_gemm<0><<<(B / 64) * (H / 32), 32, 0, stream>>>(h1, W2, b2, h2, B, H, H);
    fwd_gemm<1><<<(B / 64) * (Dout / 32), 32, 0, stream>>>(h2, W3, b3, f, B, Dout, H);

    // per-sample ||J||_F^2
    size_t shmem = (size_t)(64 * 512 + 512 + 512 + 256) * sizeof(float);
    jac_norm<<<B, 256, shmem, stream>>>(h1, h2, W1, W2, W3, ns);

    // out = softplus(K) * f / (||J|| + eps)
    finalize<<<(B * Dout + 255) / 256, 256, 0, stream>>>(f, ns, Kp, out, B * Dout);
}